// GATPairClassifier_23381801959471
// MI455X (gfx1250) — compile-verified
//
#include <hip/hip_runtime.h>
#include <hip/hip_bf16.h>

#define B_   32
#define N_   512
#define F_   128
#define D_   8
#define H_   8
#define HD_  64
#define ALPHA_ 0.2f

typedef __attribute__((ext_vector_type(2))) float v2f;
typedef __attribute__((ext_vector_type(8))) float v8f;

// ---------------------------------------------------------------------------
// K1: layer-1 feature projection, Wh1[b,n,h*8+d] = sum_f nf[b,n,f]*Ws[h,f,d]
// GEMM per batch: (512x128)@(128x64). One wave per 16x16 C tile, f32 WMMA K=4.
// A frag (16x4 f32): lane L -> row m=L&15, K = (L>>4)*2 + {0,1}  (ISA 7.12.2)
// B frag (4x16 f32): lane L -> col n=L&15, K = (L>>4)*2 + {0,1}
// C/D (16x16 f32):  lane L -> col n=L&15, VGPR r -> row r + (L>>4)*8
// ---------------------------------------------------------------------------
__global__ __launch_bounds__(128) void k1_wh1(const float* __restrict__ nf,
                                              const float* __restrict__ Ws,
                                              float* __restrict__ Wh1) {
  const int b    = blockIdx.x;
  const int m0   = blockIdx.y * 16;          // row tile in N_=512
  const int wave = threadIdx.x >> 5;         // 4 waves -> 4 N-tiles of 64 cols
  const int lane = threadIdx.x & 31;
  const int n0   = wave * 16;
  const int half = lane >> 4;
  const int mrow = lane & 15;
  const int col  = n0 + (lane & 15);
  const float* arow = nf + (size_t)(b * N_ + m0 + mrow) * F_ + half * 2;
  // Ws[h,f,d] flat = h*1024 + f*8 + d ; col = h*8+d
  const int wsbase = (col >> 3) * (F_ * D_) + (col & 7);

  v8f acc = {};
  for (int k0 = 0; k0 < F_; k0 += 4) {
    v2f a = *(const v2f*)(arow + k0);        // two consecutive K of row m
    const int kk = k0 + half * 2;
    v2f bf;
    bf.x = Ws[wsbase + kk * D_];
    bf.y = Ws[wsbase + (kk + 1) * D_];
    acc = __builtin_amdgcn_wmma_f32_16x16x4_f32(false, a, false, bf,
                                                (short)0, acc, false, false);
  }
  float* out = Wh1 + (size_t)(b * N_ + m0 + half * 8) * HD_ + col;
#pragma unroll
  for (int r = 0; r < 8; ++r) out[(size_t)r * HD_] = acc[r];
}

// ---------------------------------------------------------------------------
// K1b: attention scores. s1 = Wh.a[:8], s2 = Wh.a[8:], then the separable-exp
// factors: p = exp(s), q = exp(alpha*s).  exp(lrelu(s1+s2)) == p1*p2 if
// p1*p2>=1 else q1*q2  (since s1+s2>=0 <=> exp(s1+s2)>=1).
// ---------------------------------------------------------------------------
__global__ __launch_bounds__(256) void k1b_scores(const float* __restrict__ Wh1,
                                                  const float* __restrict__ As,
                                                  float* p1, float* q1,
                                                  float* p2, float* q2) {
  const int idx = blockIdx.x * 256 + threadIdx.x;   // B*H*N threads
  const int b = idx >> 12;
  const int h = (idx >> 9) & 7;
  const int n = idx & 511;
  const float* wh = Wh1 + (size_t)(b * N_ + n) * HD_ + h * D_;
  const float* a  = As + h * 16;
  float s1 = 0.f, s2 = 0.f;
#pragma unroll
  for (int d = 0; d < 8; ++d) { s1 += wh[d] * a[d]; s2 += wh[d] * a[8 + d]; }
  const int o = (b * H_ + h) * N_ + n;
  p1[o] = expf(s1);          q1[o] = expf(ALPHA_ * s1);
  p2[o] = expf(s2);          q2[o] = expf(ALPHA_ * s2);
}

// ---------------------------------------------------------------------------
// K2: softmax denominators (column sums over i), stored as reciprocal.
// ---------------------------------------------------------------------------
__global__ __launch_bounds__(512) void k2_denom(const float* __restrict__ p1,
                                                const float* __restrict__ q1,
                                                const float* __restrict__ p2,
                                                const float* __restrict__ q2,
                                                float* __restrict__ Dinv) {
  __shared__ float p1s[N_], q1s[N_];
  const int base = blockIdx.x * N_;   // one (b,head) per block
  const int j = threadIdx.x;
  p1s[j] = p1[base + j];  q1s[j] = q1[base + j];
  __syncthreads();
  const float pj = p2[base + j], qj = q2[base + j];
  float s = 0.f;
  for (int i = 0; i < N_; ++i) {
    const float pv = p1s[i] * pj;
    s += (pv >= 1.0f) ? pv : q1s[i] * qj;
  }
  Dinv[base + j] = 1.0f / s;
}

// ---------------------------------------------------------------------------
// K3: attention aggregation H' = E @ (Wh/D), E generated on the fly directly
// in WMMA A-fragment layout. Per (b,head): M=512, K=512, N=8 (padded to 16).
// elu applied, scattered into x[b,n,h*8+d].
// ---------------------------------------------------------------------------
__global__ __launch_bounds__(256) void k3_att(const float* __restrict__ Wh1,
                                              const float* __restrict__ p1,
                                              const float* __restrict__ q1,
                                              const float* __restrict__ p2,
                                              const float* __restrict__ q2,
                                              const float* __restrict__ Dinv,
                                              float* __restrict__ xbuf) {
  __shared__ float whn[N_ * D_];                 // Wh/D, 16 KB
  __shared__ float p1s[N_], q1s[N_], p2s[N_], q2s[N_];
  const int bh = blockIdx.x;
  const int b = bh >> 3, h = bh & 7;
  const int base = bh * N_;

  for (int j = threadIdx.x; j < N_; j += 256) {
    const float di = Dinv[base + j];
    const float* src = Wh1 + (size_t)(b * N_ + j) * HD_ + h * D_;
#pragma unroll
    for (int d = 0; d < 8; ++d) whn[j * 8 + d] = src[d] * di;
    p1s[j] = p1[base + j]; q1s[j] = q1[base + j];
    p2s[j] = p2[base + j]; q2s[j] = q2[base + j];
  }
  __syncthreads();

  const int wave = threadIdx.x >> 5, lane = threadIdx.x & 31;
  const int half = lane >> 4, n = lane & 15;
  for (int mt = wave; mt < N_ / 16; mt += 8) {      // 8 waves share 32 M-tiles
    const int i0 = mt * 16;
    const int i  = i0 + (lane & 15);
    const float p1i = p1s[i], q1i = q1s[i];
    v8f acc = {};
    for (int k0 = 0; k0 < N_; k0 += 4) {
      const int j0 = k0 + half * 2;
      v2f a, bf;
      const float e0 = p1i * p2s[j0];
      a.x = (e0 >= 1.f) ? e0 : q1i * q2s[j0];
      const float e1 = p1i * p2s[j0 + 1];
      a.y = (e1 >= 1.f) ? e1 : q1i * q2s[j0 + 1];
      bf.x = (n < 8) ? whn[j0 * 8 + n] : 0.f;       // N padded 8 -> 16
      bf.y = (n < 8) ? whn[(j0 + 1) * 8 + n] : 0.f;
      acc = __builtin_amdgcn_wmma_f32_16x16x4_f32(false, a, false, bf,
                                                  (short)0, acc, false, false);
    }
    if (n < 8) {
#pragma unroll
      for (int r = 0; r < 8; ++r) {
        const int row = i0 + r + half * 8;
        float v = acc[r];
        v = (v > 0.f) ? v : (expf(v) - 1.f);        // elu (concat head)
        xbuf[(size_t)(b * N_ + row) * HD_ + h * D_ + n] = v;
      }
    }
  }
}

// ---------------------------------------------------------------------------
// K4: layer-2 GAT (only row 0 survives) + MLP head. One block per batch.
// ---------------------------------------------------------------------------
__global__ __launch_bounds__(512) void k4_head(const float* __restrict__ xbuf,
                                               const float* __restrict__ W_out,
                                               const float* __restrict__ a_out,
                                               const float* __restrict__ feats,
                                               const float* __restrict__ lin1_w,
                                               const float* __restrict__ lin1_b,
                                               const float* __restrict__ prelu_a,
                                               const float* __restrict__ bn_gamma,
                                               const float* __restrict__ bn_beta,
                                               const float* __restrict__ lin2_w,
                                               const float* __restrict__ lin2_b,
                                               float* __restrict__ out) {
  __shared__ float wh2[N_ * D_];
  __shared__ float p1s[N_], q1s[N_], p2s[N_], q2s[N_];
  __shared__ float gat[D_];
  const int b = blockIdx.x, j = threadIdx.x;

  // phase 1: Wh2[j,:] = x[b,j,:] @ W_out  and score factors
  float w[8];
#pragma unroll
  for (int d = 0; d < 8; ++d) w[d] = 0.f;
  const float* xr = xbuf + (size_t)(b * N_ + j) * HD_;
  for (int c = 0; c < HD_; ++c) {
    const float xv = xr[c];
#pragma unroll
    for (int d = 0; d < 8; ++d) w[d] += xv * W_out[c * 8 + d];
  }
  float s1 = 0.f, s2 = 0.f;
#pragma unroll
  for (int d = 0; d < 8; ++d) {
    s1 += w[d] * a_out[d]; s2 += w[d] * a_out[8 + d]; wh2[j * 8 + d] = w[d];
  }
  p1s[j] = expf(s1); q1s[j] = expf(ALPHA_ * s1);
  p2s[j] = expf(s2); q2s[j] = expf(ALPHA_ * s2);
  __syncthreads();

  // phase 2: denom over i, weight for row i=0, per-j contribution
  const float pj = p2s[j], qj = q2s[j];
  float den = 0.f;
  for (int i = 0; i < N_; ++i) {
    const float pv = p1s[i] * pj;
    den += (pv >= 1.f) ? pv : q1s[i] * qj;
  }
  const float e0p = p1s[0] * pj;
  const float e0 = (e0p >= 1.f) ? e0p : q1s[0] * qj;
  const float t = e0 / den;
  float contrib[8];
#pragma unroll
  for (int d = 0; d < 8; ++d) contrib[d] = t * wh2[j * 8 + d];
  __syncthreads();
#pragma unroll
  for (int d = 0; d < 8; ++d) wh2[j * 8 + d] = contrib[d];

  // phase 3: tree-reduce the 512 contributions (8-vector each)
  for (int s = 256; s >= 1; s >>= 1) {
    __syncthreads();
    for (int tix = j; tix < s * 8; tix += 512) wh2[tix] += wh2[tix + s * 8];
  }
  __syncthreads();
  if (j < 8) { const float v = wh2[j]; gat[j] = (v > 0.f) ? v : (expf(v) - 1.f); }
  __syncthreads();

  // phase 4: concat -> lin1 -> prelu -> bn-scale -> lin2 (tiny, one lane)
  if (j == 0) {
    const float rbn = rsqrtf(1.0f + 1e-5f);
    float acc_o = lin2_b[0];
    for (int d = 0; d < 8; ++d) {
      float a = lin1_b[d];
      const float* wr = lin1_w + d * (F_ + D_);
      for (int c = 0; c < F_; ++c) a += feats[b * F_ + c] * wr[c];
      for (int c = 0; c < D_; ++c) a += gat[c] * wr[F_ + c];
      a = (a >= 0.f) ? a : prelu_a[0] * a;
      a = a * rbn * bn_gamma[d] + bn_beta[d];
      acc_o += a * lin2_w[d];
    }
    out[b] = acc_o;
  }
}

extern "C" void kernel_launch(void* const* d_in, const int* in_sizes, int n_in,
                              void* d_out, int out_size, void* d_ws, size_t ws_size,
                              hipStream_t stream) {
  (void)in_sizes; (void)n_in; (void)out_size; (void)ws_size;
  const float* feats  = (const float*)d_in[0];
  const float* nf     = (const float*)d_in[1];
  const float* Ws     = (const float*)d_in[2];
  const float* As     = (const float*)d_in[3];
  const float* W_out  = (const float*)d_in[4];
  const float* a_out  = (const float*)d_in[5];
  const float* lin1_w = (const float*)d_in[6];
  const float* lin1_b = (const float*)d_in[7];
  const float* prelu  = (const float*)d_in[8];
  const float* bng    = (const float*)d_in[9];
  const float* bnb    = (const float*)d_in[10];
  const float* l2w    = (const float*)d_in[11];
  const float* l2b    = (const float*)d_in[12];
  float* out = (float*)d_out;

  float* ws   = (float*)d_ws;
  float* Wh1  = ws;                               // B*N*64  = 1,048,576 f
  float* p1   = Wh1  + (size_t)B_ * N_ * HD_;     // B*H*N   =   131,072 f each
  float* q1   = p1   + (size_t)B_ * H_ * N_;
  float* p2   = q1   + (size_t)B_ * H_ * N_;
  float* q2   = p2   + (size_t)B_ * H_ * N_;
  float* Dinv = q2   + (size_t)B_ * H_ * N_;
  float* xbuf = Dinv + (size_t)B_ * H_ * N_;      // B*N*64

  k1_wh1    <<<dim3(B_, N_ / 16), 128, 0, stream>>>(nf, Ws, Wh1);
  k1b_scores<<<(B_ * H_ * N_) / 256, 256, 0, stream>>>(Wh1, As, p1, q1, p2, q2);
  k2_denom  <<<B_ * H_, N_, 0, stream>>>(p1, q1, p2, q2, Dinv);
  k3_att    <<<B_ * H_, 256, 0, stream>>>(Wh1, p1, q1, p2, q2, Dinv, xbuf);
  k4_head   <<<B_, N_, 0, stream>>>(xbuf, W_out, a_out, feats, lin1_w, lin1_b,
                                    prelu, bng, bnb, l2w, l2b, out);
}